// FastGuidedFilter_35931696398901
// MI455X (gfx1250) — compile-verified
//
#include <hip/hip_runtime.h>

typedef float v2f __attribute__((ext_vector_type(2)));
typedef float v4f __attribute__((ext_vector_type(4)));
typedef float v8f __attribute__((ext_vector_type(8)));

#define LRW 512
#define HRW 2048
#define NPLANES 12
#define EPSV 0.01f

// ---------------------------------------------------------------------------
// Phase 1: fused low-res guided-filter solve.
// One wave (32 lanes) per 16x16 output tile of one plane.
// Loads a 20x20 clamped halo of x,y; box-filters x,y,x*x,x*y (separable, LDS);
// computes A,b on an 18x18 halo region; final 3x3 box filter of A and b is
// done as (vertical 3-tap VALU sum) x (banded 1/9 matrix) via
// V_WMMA_F32_16X16X4_F32 (5 chained K=4 steps over K=20 zero-padded).
// ---------------------------------------------------------------------------
__global__ __launch_bounds__(32) void gf_lowres_kernel(
    const float* __restrict__ x_lr, const float* __restrict__ y_lr,
    float* __restrict__ meanA, float* __restrict__ meanB)
{
    const int tx = blockIdx.x;      // tile col  (0..31)
    const int ty = blockIdx.y;      // tile row  (0..31)
    const int plane = blockIdx.z;   // 0..11
    const long po = (long)plane * LRW * LRW;
    const float* __restrict__ xp = x_lr + po;
    const float* __restrict__ yp = y_lr + po;

    __shared__ float xs[20][21], ys[20][21];
    __shared__ float svx[18][21], svy[18][21], svxx[18][21], svxy[18][21];
    __shared__ float A18[18][21], B18[18][21];
    __shared__ float Ta[16][20], Tb[16][20];

    const int lane = threadIdx.x;        // 0..31
    const int r0 = ty * 16 - 2;
    const int c0 = tx * 16 - 2;
    const float inv9 = 1.0f / 9.0f;

    // 20x20 clamped halo load (edge-replication pad == clamped indexing)
    for (int idx = lane; idx < 400; idx += 32) {
        int r = idx / 20, c = idx % 20;
        int gr = min(max(r0 + r, 0), LRW - 1);
        int gc = min(max(c0 + c, 0), LRW - 1);
        xs[r][c] = xp[gr * LRW + gc];
        ys[r][c] = yp[gr * LRW + gc];
    }
    __syncthreads();

    // vertical 3-tap sums of x, y, x*x, x*y -> 18x20
    for (int idx = lane; idx < 360; idx += 32) {
        int r = idx / 20, c = idx % 20;
        float x0 = xs[r][c], x1 = xs[r + 1][c], x2 = xs[r + 2][c];
        float y0 = ys[r][c], y1 = ys[r + 1][c], y2 = ys[r + 2][c];
        svx[r][c]  = x0 + x1 + x2;
        svy[r][c]  = y0 + y1 + y2;
        svxx[r][c] = x0 * x0 + x1 * x1 + x2 * x2;
        svxy[r][c] = x0 * y0 + x1 * y1 + x2 * y2;
    }
    __syncthreads();

    // horizontal pass -> means on 18x18; solve A, b
    for (int idx = lane; idx < 324; idx += 32) {
        int r = idx / 18, c = idx % 18;
        float mx  = (svx[r][c]  + svx[r][c + 1]  + svx[r][c + 2])  * inv9;
        float my  = (svy[r][c]  + svy[r][c + 1]  + svy[r][c + 2])  * inv9;
        float mxx = (svxx[r][c] + svxx[r][c + 1] + svxx[r][c + 2]) * inv9;
        float mxy = (svxy[r][c] + svxy[r][c + 1] + svxy[r][c + 2]) * inv9;
        float cov = mxy - mx * my;
        float var = mxx - mx * mx;
        float a = cov / (var + EPSV);
        A18[r][c] = a;
        B18[r][c] = my - a * mx;
    }
    __syncthreads();

    // vertical 3-tap sum of A, b -> 16x18 (cols 18,19 zero pad for K=20)
    for (int idx = lane; idx < 320; idx += 32) {
        int r = idx / 20, c = idx % 20;
        float ta = 0.0f, tb = 0.0f;
        if (c < 18) {
            ta = A18[r][c] + A18[r + 1][c] + A18[r + 2][c];
            tb = B18[r][c] + B18[r + 1][c] + B18[r + 2][c];
        }
        Ta[r][c] = ta;
        Tb[r][c] = tb;
    }
    __syncthreads();

    // Horizontal band filter as matmul: D = T(16x20) x Wh(20x16), Wh banded 1/9.
    // A f32 16x4 layout: lanes0-15 VGPR0=K0,VGPR1=K1; lanes16-31 VGPR0=K2,VGPR1=K3.
    // B f32 4x16 layout: lane = N; lanes0-15 rows K0/K1, lanes16-31 rows K2/K3.
    const int m = lane & 15;
    const int half = lane >> 4;
    v8f accA = {};
    v8f accB = {};
#pragma unroll
    for (int kb = 0; kb < 20; kb += 4) {
        const int k0 = kb + 2 * half;
        const int k1 = k0 + 1;
        v2f fa, fb, wB;
        fa.x = Ta[m][k0]; fa.y = Ta[m][k1];
        fb.x = Tb[m][k0]; fb.y = Tb[m][k1];
        // Wh[k][n] = 1/9 iff n <= k <= n+2 and k < 18
        wB.x = (k0 < 18 && m <= k0 && k0 <= m + 2) ? inv9 : 0.0f;
        wB.y = (k1 < 18 && m <= k1 && k1 <= m + 2) ? inv9 : 0.0f;
        accA = __builtin_amdgcn_wmma_f32_16x16x4_f32(
            false, fa, false, wB, (short)0, accA, false, false);
        accB = __builtin_amdgcn_wmma_f32_16x16x4_f32(
            false, fb, false, wB, (short)0, accB, false, false);
    }

    // D 16x16 layout: VGPR v, lanes0-15 -> M=v; lanes16-31 -> M=v+8; N=lane&15
    float* __restrict__ mAo = meanA + po;
    float* __restrict__ mBo = meanB + po;
    const int colO = tx * 16 + m;
#pragma unroll
    for (int v = 0; v < 8; v++) {
        int rowO = ty * 16 + v + 8 * half;
        mAo[rowO * LRW + colO] = accA[v];
        mBo[rowO * LRW + colO] = accB[v];
    }
}

// ---------------------------------------------------------------------------
// Phase 2: streaming HR pass. Each block: 256 threads x 4 px = 1024 contiguous
// pixels of one row. The two LR source rows (i0,i1) of mean_A/mean_b are
// staged into LDS (<=260 cols cover the block's window at scale 511/2047).
// x_hr read and output written with nontemporal float4 (pure streaming).
// ---------------------------------------------------------------------------
__global__ __launch_bounds__(256) void gf_upsample_kernel(
    const float* __restrict__ x_hr, const float* __restrict__ meanA,
    const float* __restrict__ meanB, float* __restrict__ out)
{
    const float S = 511.0f / 2047.0f;   // align_corners scale
    const int seg = blockIdx.x;         // 0..1
    const int row = blockIdx.y;         // 0..2047
    const int plane = blockIdx.z;       // 0..11

    // vertical interp (uniform per block)
    const float py = (float)row * S;
    int i0 = (int)py;
    if (i0 > 511) i0 = 511;
    const int i1 = min(i0 + 1, 511);
    const float wy = py - (float)i0;

    const int j0b = seg * 1024;
    const int cbase = (int)((float)j0b * S);

    __shared__ float sA0[260], sA1[260], sB0[260], sB1[260];
    const long po = (long)plane * LRW * LRW;
    const float* __restrict__ Ap = meanA + po;
    const float* __restrict__ Bp = meanB + po;
    for (int c = threadIdx.x; c < 260; c += 256) {
        int gc = min(cbase + c, 511);
        sA0[c] = Ap[i0 * LRW + gc];
        sA1[c] = Ap[i1 * LRW + gc];
        sB0[c] = Bp[i0 * LRW + gc];
        sB1[c] = Bp[i1 * LRW + gc];
    }
    __syncthreads();

    const long rowOff = ((long)plane * HRW + row) * HRW + j0b;
    const v4f* __restrict__ xin = (const v4f*)(x_hr + rowOff);
    v4f* __restrict__ op = (v4f*)(out + rowOff);
    const int t = threadIdx.x;

    v4f xv = __builtin_nontemporal_load(&xin[t]);
    v4f rv;
#pragma unroll
    for (int u = 0; u < 4; u++) {
        int j = j0b + t * 4 + u;
        float px = (float)j * S;
        int c0 = (int)px;
        float wx = px - (float)c0;
        int l0 = c0 - cbase;
        int l1 = l0 + 1;            // staged with clamp; l1 <= 257 < 260
        float a0 = sA0[l0] + wx * (sA0[l1] - sA0[l0]);
        float a1 = sA1[l0] + wx * (sA1[l1] - sA1[l0]);
        float mA = a0 + wy * (a1 - a0);
        float b0 = sB0[l0] + wx * (sB0[l1] - sB0[l0]);
        float b1 = sB1[l0] + wx * (sB1[l1] - sB1[l0]);
        float mB = b0 + wy * (b1 - b0);
        float r = fmaf(mA, xv[u], mB);
        rv[u] = fminf(fmaxf(r, 0.0f), 255.0f);
    }
    __builtin_nontemporal_store(rv, &op[t]);
}

extern "C" void kernel_launch(void* const* d_in, const int* in_sizes, int n_in,
                              void* d_out, int out_size, void* d_ws, size_t ws_size,
                              hipStream_t stream) {
    (void)in_sizes; (void)n_in; (void)out_size; (void)ws_size;
    const float* x_lr = (const float*)d_in[0];
    const float* y_lr = (const float*)d_in[1];
    const float* x_hr = (const float*)d_in[2];
    float* out = (float*)d_out;

    float* meanA = (float*)d_ws;                       // 12*512*512 floats
    float* meanB = meanA + (long)NPLANES * LRW * LRW;  // 12*512*512 floats

    dim3 g1(LRW / 16, LRW / 16, NPLANES);
    gf_lowres_kernel<<<g1, dim3(32), 0, stream>>>(x_lr, y_lr, meanA, meanB);

    dim3 g2(2, HRW, NPLANES);
    gf_upsample_kernel<<<g2, dim3(256), 0, stream>>>(x_hr, meanA, meanB, out);
}